// DecoderRNN_67714454389121
// MI455X (gfx1250) — compile-verified
//
#include <hip/hip_runtime.h>
#include <math.h>

typedef float v2f __attribute__((ext_vector_type(2)));
typedef float v8f __attribute__((ext_vector_type(8)));

#define Bx   32
#define Tt   64
#define TM1  63
#define Sx   128
#define Hx   512
#define Ex   512
#define Vx   32000
#define VE   32032   // V + OOV

// ---------------- WMMA helpers (f32 16x16x4, full f32 precision) ----------------
__device__ __forceinline__ v8f wmma4(v2f a, v2f b, v8f c) {
  // (neg_a, A, neg_b, B, c_mod, C, reuse_a, reuse_b)
  return __builtin_amdgcn_wmma_f32_16x16x4_f32(false, a, false, b, (short)0, c,
                                               false, false);
}

// C(16x16) += A[arow0:+16, 0:K] * W[wrow0:+16, 0:K]^T   (i.e. X @ W.T tile)
// A row-major lda, W row-major ldw. Lane l: row/col = l&15, K-offset = (l>>4)*2.
// K must be a compile-time constant at each call site (constant trip count).
__device__ __forceinline__ void gemm_tile(const float* __restrict__ A, int lda, int arow0,
                                          const float* __restrict__ W, int ldw, int wrow0,
                                          int K, v8f& acc) {
  const int l  = threadIdx.x & 31;
  const int n  = l & 15;
  const int ko = (l >> 4) * 2;
  const float* ap = A + (size_t)(arow0 + n) * lda + ko;
  const float* wp = W + (size_t)(wrow0 + n) * ldw + ko;
#pragma unroll 4
  for (int k = 0; k < K; k += 4) {
    v2f a = *(const v2f*)(ap + k);
    v2f b = *(const v2f*)(wp + k);
    acc = wmma4(a, b, acc);
  }
}

// order-preserving float<->uint for atomicMax on floats
__device__ __forceinline__ unsigned enc_f32(float f) {
  unsigned u = __float_as_uint(f);
  return (u & 0x80000000u) ? ~u : (u | 0x80000000u);
}
__device__ __forceinline__ float dec_f32(unsigned u) {
  unsigned v = (u & 0x80000000u) ? (u & 0x7FFFFFFFu) : ~u;
  return __uint_as_float(v);
}

// ---------------- kernels ----------------
__global__ void init_kernel(const float* __restrict__ h0, float* __restrict__ h,
                            float* __restrict__ c) {
  int i = blockIdx.x * blockDim.x + threadIdx.x;
  if (i < Bx * Hx) { h[i] = h0[i]; c[i] = 0.f; }
}

// x = [emb_W[tok], last_out]  (32 x 1024)
__global__ void emb_kernel(const int* __restrict__ targets, int t,
                           const float* __restrict__ emb_W,
                           const float* __restrict__ last_out, float* __restrict__ x) {
  int b = blockIdx.x;
  int tok = targets[b * Tt + t];
  for (int k = threadIdx.x; k < Ex + Hx; k += blockDim.x)
    x[b * (Ex + Hx) + k] = (k < Ex) ? emb_W[(size_t)tok * Ex + k]
                                    : last_out[b * Hx + (k - Ex)];
}

// GRU cell: 32 H-chunks; 6 waves/WG, each owns one (gate, mtile) and computes
// BOTH the gi (K=1024) and gh (K=512) tiles with constant-trip-count loops.
// No wave-divergent control flow anywhere near the WMMAs.
__global__ void gru_kernel(const float* __restrict__ x, const float* __restrict__ h_in,
                           const float* __restrict__ W_ih, const float* __restrict__ W_hh,
                           const float* __restrict__ b_ih, const float* __restrict__ b_hh,
                           float* __restrict__ h_out) {
  __shared__ float tile[12][16][16];
  const int wave = threadIdx.x >> 5;           // 0..5
  const int j    = blockIdx.x;                 // H column chunk (16 cols)
  const int gate = wave >> 1;                  // 0:r 1:z 2:n
  const int mt   = wave & 1;                   // M tile (rows 0-15 / 16-31)
  v8f gi = {}, gh = {};
  gemm_tile(x,    Ex + Hx, mt * 16, W_ih, Ex + Hx, gate * Hx + j * 16, Ex + Hx, gi);
  gemm_tile(h_in, Hx,      mt * 16, W_hh, Hx,      gate * Hx + j * 16, Hx,      gh);
  const int l = threadIdx.x & 31;
  const int n = l & 15, rb = (l >> 4) * 8;
#pragma unroll
  for (int r = 0; r < 8; ++r) {
    tile[wave][rb + r][n]     = gi[r];
    tile[6 + wave][rb + r][n] = gh[r];
  }
  __syncthreads();
  for (int idx = threadIdx.x; idx < Bx * 16; idx += blockDim.x) {
    int m = idx >> 4, c = idx & 15;
    int mtt = m >> 4, mm = m & 15;
    int col = j * 16 + c;
    float ir = tile[0 + mtt][mm][c]  + b_ih[col];
    float iz = tile[2 + mtt][mm][c]  + b_ih[Hx + col];
    float in_= tile[4 + mtt][mm][c]  + b_ih[2 * Hx + col];
    float hr = tile[6 + mtt][mm][c]  + b_hh[col];
    float hz = tile[8 + mtt][mm][c]  + b_hh[Hx + col];
    float hn = tile[10 + mtt][mm][c] + b_hh[2 * Hx + col];
    float rr = 1.f / (1.f + expf(-(ir + hr)));
    float zz = 1.f / (1.f + expf(-(iz + hz)));
    float nn = tanhf(in_ + rr * hn);
    h_out[m * Hx + col] = (1.f - zz) * nn + zz * h_in[m * Hx + col];
  }
}

// batched dot attention + masked softmax + context (per-batch B matrix -> VALU)
__global__ void attn_kernel(const float* __restrict__ rnn, const float* __restrict__ enc,
                            const int* __restrict__ lens,
                            float* __restrict__ attn, float* __restrict__ ctx) {
  int b = blockIdx.x, s = threadIdx.x;
  __shared__ float red[Sx];
  __shared__ float aw[Sx];
  const float* e = enc + ((size_t)b * Sx + s) * Hx;
  const float* r = rnn + b * Hx;
  float dot = 0.f;
#pragma unroll 4
  for (int k = 0; k < Hx; ++k) dot += r[k] * e[k];
  if (s >= lens[b]) dot = -1e9f;
  red[s] = dot; __syncthreads();
  for (int off = 64; off; off >>= 1) {
    if (s < off) red[s] = fmaxf(red[s], red[s + off]);
    __syncthreads();
  }
  float mx = red[0]; __syncthreads();
  float ex = expf(dot - mx);
  red[s] = ex; __syncthreads();
  for (int off = 64; off; off >>= 1) {
    if (s < off) red[s] += red[s + off];
    __syncthreads();
  }
  float a = ex / red[0];
  attn[b * Sx + s] = a;
  aw[s] = a; __syncthreads();
  for (int h = s; h < Hx; h += Sx) {
    float acc = 0.f;
    const float* eb = enc + (size_t)b * Sx * Hx + h;
    for (int ss = 0; ss < Sx; ++ss) acc += aw[ss] * eb[ss * Hx];
    ctx[b * Hx + h] = acc;
  }
}

// concat_out = tanh([rnn, ctx] @ W_c.T + b_c); 2 M-tiles x 2 N-tiles per wave
__global__ void wc_kernel(const float* __restrict__ rnn, const float* __restrict__ ctx,
                          const float* __restrict__ W_c, const float* __restrict__ b_c,
                          float* __restrict__ cout) {
  const int wave = threadIdx.x >> 5;
  const int jtp = blockIdx.x * 4 + wave;       // 0..15, owns cols jtp*32..+31
  const int l = threadIdx.x & 31;
  const int n = l & 15, ko = (l >> 4) * 2;
  const float* wp0 = W_c + (size_t)(jtp * 32 + n) * (2 * Hx) + ko;
  const float* wp1 = W_c + (size_t)(jtp * 32 + 16 + n) * (2 * Hx) + ko;
  v8f a00 = {}, a10 = {}, a01 = {}, a11 = {};
#pragma unroll 4
  for (int k = 0; k < Hx; k += 4) {            // K = 0..511 from rnn
    v2f b0 = *(const v2f*)(wp0 + k);
    v2f b1 = *(const v2f*)(wp1 + k);
    v2f x0 = *(const v2f*)(rnn + n * Hx + ko + k);
    v2f x1 = *(const v2f*)(rnn + (16 + n) * Hx + ko + k);
    a00 = wmma4(x0, b0, a00);
    a10 = wmma4(x1, b0, a10);
    a01 = wmma4(x0, b1, a01);
    a11 = wmma4(x1, b1, a11);
  }
#pragma unroll 4
  for (int k = 0; k < Hx; k += 4) {            // K = 512..1023 from ctx
    v2f b0 = *(const v2f*)(wp0 + Hx + k);
    v2f b1 = *(const v2f*)(wp1 + Hx + k);
    v2f x0 = *(const v2f*)(ctx + n * Hx + ko + k);
    v2f x1 = *(const v2f*)(ctx + (16 + n) * Hx + ko + k);
    a00 = wmma4(x0, b0, a00);
    a10 = wmma4(x1, b0, a10);
    a01 = wmma4(x0, b1, a01);
    a11 = wmma4(x1, b1, a11);
  }
  float bias0 = b_c[jtp * 32 + n];
  float bias1 = b_c[jtp * 32 + 16 + n];
  int rb = (l >> 4) * 8;
#pragma unroll
  for (int r = 0; r < 8; ++r) {
    cout[(rb + r) * Hx + jtp * 32 + n]           = tanhf(a00[r] + bias0);
    cout[(16 + rb + r) * Hx + jtp * 32 + n]      = tanhf(a10[r] + bias0);
    cout[(rb + r) * Hx + jtp * 32 + 16 + n]      = tanhf(a01[r] + bias1);
    cout[(16 + rb + r) * Hx + jtp * 32 + 16 + n] = tanhf(a11[r] + bias1);
  }
}

// p_gen = sigmoid([ctx, rnn, emb] @ W_p.T + b_p); also resets per-step softmax state
__global__ void pgen_kernel(const float* __restrict__ ctx, const float* __restrict__ rnn,
                            const float* __restrict__ x, const float* __restrict__ W_p,
                            const float* __restrict__ b_p, float* __restrict__ pgen,
                            unsigned* __restrict__ rmax, float* __restrict__ rsum,
                            float* __restrict__ out_pg) {
  int b = blockIdx.x, tid = threadIdx.x;
  __shared__ float red[64];
  float acc = 0.f;
  for (int k = tid; k < 2 * Hx + Ex; k += 64) {
    float v;
    if (k < Hx)            v = ctx[b * Hx + k];
    else if (k < 2 * Hx)   v = rnn[b * Hx + (k - Hx)];
    else                   v = x[b * (Ex + Hx) + (k - 2 * Hx)];  // input_emb
    acc += v * W_p[k];
  }
  red[tid] = acc; __syncthreads();
  for (int off = 32; off; off >>= 1) {
    if (tid < off) red[tid] += red[tid + off];
    __syncthreads();
  }
  if (tid == 0) {
    float pg = 1.f / (1.f + expf(-(red[0] + b_p[0])));
    pgen[b] = pg;
    out_pg[b] = pg;
    rmax[b] = 0u;      // below enc of any finite float
    rsum[b] = 0.f;
  }
}

// logits = concat @ W_o.T + b_o  (32x32000) — dominant GEMM.
// 2 M-tiles x 2 N-tiles per wave: 4 b64 loads -> 4 WMMAs per K-step.
__global__ void vocab_kernel(const float* __restrict__ cc, const float* __restrict__ W_o,
                             const float* __restrict__ b_o, float* __restrict__ logits) {
  const int wave = threadIdx.x >> 5;
  const int jtp = blockIdx.x * 4 + wave;       // 0..999, owns vocab cols jtp*32..+31
  const int l = threadIdx.x & 31;
  const int n = l & 15, ko = (l >> 4) * 2;
  const float* wp0 = W_o + (size_t)(jtp * 32 + n) * Hx + ko;
  const float* wp1 = W_o + (size_t)(jtp * 32 + 16 + n) * Hx + ko;
  const float* a0p = cc + n * Hx + ko;
  const float* a1p = cc + (16 + n) * Hx + ko;
  __builtin_prefetch(wp0, 0, 0);
  __builtin_prefetch(wp1, 0, 0);
  v8f a00 = {}, a10 = {}, a01 = {}, a11 = {};
#pragma unroll 4
  for (int k = 0; k < Hx; k += 4) {
    v2f b0 = *(const v2f*)(wp0 + k);
    v2f b1 = *(const v2f*)(wp1 + k);
    v2f x0 = *(const v2f*)(a0p + k);
    v2f x1 = *(const v2f*)(a1p + k);
    a00 = wmma4(x0, b0, a00);
    a10 = wmma4(x1, b0, a10);
    a01 = wmma4(x0, b1, a01);
    a11 = wmma4(x1, b1, a11);
  }
  float bias0 = b_o[jtp * 32 + n];
  float bias1 = b_o[jtp * 32 + 16 + n];
  int rb = (l >> 4) * 8;
  float* lg0 = logits + (size_t)jtp * 32 + n;
  float* lg1 = lg0 + 16;
#pragma unroll
  for (int r = 0; r < 8; ++r) {
    lg0[(size_t)(rb + r) * Vx]      = a00[r] + bias0;
    lg0[(size_t)(16 + rb + r) * Vx] = a10[r] + bias0;
    lg1[(size_t)(rb + r) * Vx]      = a01[r] + bias1;
    lg1[(size_t)(16 + rb + r) * Vx] = a11[r] + bias1;
  }
}

__global__ void rowmax_kernel(const float* __restrict__ logits, unsigned* __restrict__ rmax) {
  int b = blockIdx.x >> 3;
  int base = (blockIdx.x & 7) * 4000;
  __shared__ float red[256];
  float m = -3.0e38f;
  for (int c = threadIdx.x; c < 4000; c += 256)
    m = fmaxf(m, logits[(size_t)b * Vx + base + c]);
  red[threadIdx.x] = m; __syncthreads();
  for (int off = 128; off; off >>= 1) {
    if (threadIdx.x < off) red[threadIdx.x] = fmaxf(red[threadIdx.x], red[threadIdx.x + off]);
    __syncthreads();
  }
  if (threadIdx.x == 0) atomicMax(&rmax[b], enc_f32(red[0]));
}

__global__ void expsum_kernel(float* __restrict__ logits, const unsigned* __restrict__ rmax,
                              float* __restrict__ rsum) {
  int b = blockIdx.x >> 3;
  int base = (blockIdx.x & 7) * 4000;
  float m = dec_f32(rmax[b]);
  __shared__ float red[256];
  float s = 0.f;
  for (int c = threadIdx.x; c < 4000; c += 256) {
    size_t i = (size_t)b * Vx + base + c;
    float e = expf(logits[i] - m);
    logits[i] = e;
    s += e;
  }
  red[threadIdx.x] = s; __syncthreads();
  for (int off = 128; off; off >>= 1) {
    if (threadIdx.x < off) red[threadIdx.x] += red[threadIdx.x + off];
    __syncthreads();
  }
  if (threadIdx.x == 0) atomicAdd(&rsum[b], red[0]);
}

// p_ext base: p_gen * softmax(logits), zeros in OOV tail.
// Non-temporal stores: keep the 258 MB output stream from evicting W_o in L2.
__global__ void final_kernel(const float* __restrict__ logits, const float* __restrict__ rsum,
                             const float* __restrict__ pgen, float* __restrict__ outp) {
  int b = blockIdx.y;
  int col = blockIdx.x * 256 + threadIdx.x;
  if (col >= VE) return;
  float inv = pgen[b] / rsum[b];
  float v = 0.f;
  if (col < Vx) v = inv * __builtin_nontemporal_load(&logits[(size_t)b * Vx + col]);
  __builtin_nontemporal_store(v, &outp[(size_t)b * VE + col]);
}

// scatter-add copy probabilities
__global__ void scatter_kernel(const int* __restrict__ inputs, const float* __restrict__ attn,
                               const float* __restrict__ pgen, float* __restrict__ outp) {
  int b = blockIdx.x, s = threadIdx.x;
  float pc = (1.f - pgen[b]) * attn[b * Sx + s];
  atomicAdd(&outp[(size_t)b * VE + inputs[b * Sx + s]], pc);
}

// ---------------- host orchestration ----------------
extern "C" void kernel_launch(void* const* d_in, const int* in_sizes, int n_in,
                              void* d_out, int out_size, void* d_ws, size_t ws_size,
                              hipStream_t stream) {
  (void)in_sizes; (void)n_in; (void)out_size; (void)ws_size;
  const int*   targets = (const int*)d_in[0];
  const float* h0      = (const float*)d_in[1];
  const float* enc     = (const float*)d_in[2];
  const int*   inputs  = (const int*)d_in[3];
  const int*   lens    = (const int*)d_in[4];
  const float* emb_W   = (const float*)d_in[6];
  const float* W_ih    = (const float*)d_in[7];
  const float* W_hh    = (const float*)d_in[8];
  const float* b_ih    = (const float*)d_in[9];
  const float* b_hh    = (const float*)d_in[10];
  const float* W_c     = (const float*)d_in[11];
  const float* b_c     = (const float*)d_in[12];
  const float* W_o     = (const float*)d_in[13];
  const float* b_o     = (const float*)d_in[14];
  const float* W_p     = (const float*)d_in[15];
  const float* b_p     = (const float*)d_in[16];
  float* out = (float*)d_out;

  float* ws   = (float*)d_ws;
  float* hA   = ws;                 // 32*512
  float* hB   = hA + Bx * Hx;
  float* cA   = hB + Bx * Hx;
  float* cB   = cA + Bx * Hx;
  float* x    = cB + Bx * Hx;       // 32*1024
  float* attn = x + Bx * (Ex + Hx); // 32*128
  float* ctx  = attn + Bx * Sx;     // 32*512
  float* pgen = ctx + Bx * Hx;      // 32
  unsigned* rmax = (unsigned*)(pgen + Bx);
  float* rsum = pgen + 2 * Bx;
  float* logits = pgen + 3 * Bx;    // 32*32000
  const size_t PG_OFF = (size_t)TM1 * Bx * VE;

  init_kernel<<<(Bx * Hx + 255) / 256, 256, 0, stream>>>(h0, hA, cA);
  float* hb[2] = {hA, hB};
  float* cb[2] = {cA, cB};
  int cur = 0;
  for (int t = 0; t < TM1; ++t) {
    float* h_in = hb[cur];  float* h_out = hb[cur ^ 1];
    float* c_in = cb[cur];  float* c_out = cb[cur ^ 1];
    float* outp = out + (size_t)t * Bx * VE;
    emb_kernel   <<<Bx, 256, 0, stream>>>(targets, t, emb_W, c_in, x);
    gru_kernel   <<<Hx / 16, 192, 0, stream>>>(x, h_in, W_ih, W_hh, b_ih, b_hh, h_out);
    attn_kernel  <<<Bx, Sx, 0, stream>>>(h_out, enc, lens, attn, ctx);
    wc_kernel    <<<4, 128, 0, stream>>>(h_out, ctx, W_c, b_c, c_out);
    pgen_kernel  <<<Bx, 64, 0, stream>>>(ctx, h_out, x, W_p, b_p, pgen, rmax, rsum,
                                         out + PG_OFF + (size_t)t * Bx);
    vocab_kernel <<<250, 128, 0, stream>>>(c_out, W_o, b_o, logits);
    rowmax_kernel<<<Bx * 8, 256, 0, stream>>>(logits, rmax);
    expsum_kernel<<<Bx * 8, 256, 0, stream>>>(logits, rmax, rsum);
    final_kernel <<<dim3((VE + 255) / 256, Bx), 256, 0, stream>>>(logits, rsum, pgen, outp);
    scatter_kernel<<<Bx, Sx, 0, stream>>>(inputs, attn, pgen, outp);
    cur ^= 1;
  }
}